// InterAgg_5755256177390
// MI455X (gfx1250) — compile-verified
//
#include <hip/hip_runtime.h>
#include <hip/hip_bf16.h>

// ---------------------------------------------------------------------------
// CDNA5 (gfx1250) implementation.
//  * AGG_WEIGHT[0]==0 kills the whole neighbor-filter/top-k branch -> skipped.
//  * GEMMs use V_WMMA_F32_16X16X4_F32 (full f32 accuracy; GEMM FLOPs are tiny
//    vs the L2-bound edge-scatter phase, so fp32 WMMA is the right precision).
//  * 6 Laplacian applications: prescaled float2 gather (one 256B segment per
//    wave) + global_atomic_add_f32 into an L2-resident 12.8MB accumulator.
//  * agg reset is fused into the lap/poly consumers (zeroed once up front).
//  * W3@weight folded into a 64x64 matrix so the 50000-row W3 projection
//    collapses to 8192 gathered rows.
// Workspace requirement: ~52 MB.
// ---------------------------------------------------------------------------

typedef __attribute__((ext_vector_type(2))) float v2f;
typedef __attribute__((ext_vector_type(8))) float v8f;

#define FDIM 128
#define EDIM 64

__device__ __forceinline__ v8f v8zero() {
  v8f z = {0.f, 0.f, 0.f, 0.f, 0.f, 0.f, 0.f, 0.f};
  return z;
}

// D = A(16x4,f32) * B(4x16,f32) + C(16x16,f32)
__device__ __forceinline__ v8f wmma4(v2f a, v2f b, v8f c) {
  return __builtin_amdgcn_wmma_f32_16x16x4_f32(false, a, false, b, (short)0, c,
                                               false, false);
}

// ---------------------------------------------------------------------------
// small utility kernels
// ---------------------------------------------------------------------------
__global__ void k_zero(float* __restrict__ p, int n) {
  int i = blockIdx.x * blockDim.x + threadIdx.x;
  if (i < n) p[i] = 0.f;
}

__global__ void k_zero4(float4* __restrict__ p, int n4) {
  int i = blockIdx.x * blockDim.x + threadIdx.x;
  if (i < n4) p[i] = make_float4(0.f, 0.f, 0.f, 0.f);
}

__global__ void k_deg(const int* __restrict__ dst, float* __restrict__ deg, int E) {
  int i = blockIdx.x * blockDim.x + threadIdx.x;
  if (i < E) atomicAdd(&deg[dst[i]], 1.0f);
}

__global__ void k_dinv(float* __restrict__ deg, int N) {
  int i = blockIdx.x * blockDim.x + threadIdx.x;
  if (i < N) deg[i] = rsqrtf(fmaxf(deg[i], 1.0f));
}

// W3W = W3[64x128] @ weight[128x64]; b3w = b3[128] @ weight[128x64]
__global__ void k_w3w(const float* __restrict__ W3, const float* __restrict__ weight,
                      const float* __restrict__ b3, float* __restrict__ W3W,
                      float* __restrict__ b3w) {
  int t = threadIdx.x;  // 256 threads, 1 block
  for (int o = t; o < EDIM * EDIM; o += 256) {
    int r = o >> 6, c = o & 63;
    float s = 0.f;
    for (int k = 0; k < FDIM; ++k) s += W3[r * FDIM + k] * weight[k * EDIM + c];
    W3W[o] = s;
  }
  if (t < EDIM) {
    float s = 0.f;
    for (int k = 0; k < FDIM; ++k) s += b3[k] * weight[k * EDIM + t];
    b3w[t] = s;
  }
}

// center_scores[i] = feat[nodes[i]] @ Wclf + bclf  (one wave32 per node)
__global__ void __launch_bounds__(256) k_scores(
    const float* __restrict__ feat, const float* __restrict__ Wclf,
    const float* __restrict__ bclf, const int* __restrict__ nodes,
    float* __restrict__ out_scores, int n) {
  int wid = (blockIdx.x * blockDim.x + threadIdx.x) >> 5;
  int lane = threadIdx.x & 31;
  if (wid >= n) return;
  const float* row = feat + (size_t)nodes[wid] * FDIM;
  float s0 = 0.f, s1 = 0.f;
  for (int j = lane; j < FDIM; j += 32) {
    float f = row[j];
    s0 += f * Wclf[j * 2 + 0];
    s1 += f * Wclf[j * 2 + 1];
  }
  for (int off = 16; off; off >>= 1) {
    s0 += __shfl_xor(s0, off, 32);
    s1 += __shfl_xor(s1, off, 32);
  }
  if (lane == 0) {
    out_scores[wid * 2 + 0] = s0 + bclf[0];
    out_scores[wid * 2 + 1] = s1 + bclf[1];
  }
}

// ---------------------------------------------------------------------------
// fused MLP: h = relu(relu(feat@W1+b1)@W2+b2), also writes s = h * dinv[row].
// One wave32 per 16-row tile; A staged in LDS; C held as v8f per N-tile.
// ---------------------------------------------------------------------------
__global__ void __launch_bounds__(128) k_mlp(
    const float* __restrict__ feat, const float* __restrict__ W1,
    const float* __restrict__ b1, const float* __restrict__ W2,
    const float* __restrict__ b2, const float* __restrict__ dinv,
    float* __restrict__ hout, float* __restrict__ sout, int Nrows) {
  __shared__ __align__(16) float At[4][16][FDIM];  // 32 KB
  const int wave = threadIdx.x >> 5;
  const int lane = threadIdx.x & 31;
  const int half = lane >> 4;  // which 16-lane half
  const int l15 = lane & 15;
  const int m0 = (blockIdx.x * 4 + wave) * 16;
  const bool active = (m0 < Nrows);  // Nrows is a multiple of 16 (50000)

  if (active) {
    const float4* base = (const float4*)(feat + (size_t)m0 * FDIM);
#pragma unroll
    for (int i = 0; i < 16; ++i) {  // 512 float4 per tile, coalesced
      int idx = i * 32 + lane;
      int r = idx >> 5, c4 = idx & 31;
      *(float4*)&At[wave][r][c4 * 4] = base[r * 32 + c4];
    }
  }
  __syncthreads();
  if (!active) return;

  // ---- layer 1: [16x128] @ [128x64] ----
  v8f acc[4] = {v8zero(), v8zero(), v8zero(), v8zero()};
  for (int k0 = 0; k0 < FDIM; k0 += 4) {
    const int ka = k0 + 2 * half;
    v2f a;
    a.x = At[wave][l15][ka];
    a.y = At[wave][l15][ka + 1];
#pragma unroll
    for (int nt = 0; nt < 4; ++nt) {
      v2f b;
      b.x = W1[ka * EDIM + nt * 16 + l15];
      b.y = W1[(ka + 1) * EDIM + nt * 16 + l15];
      acc[nt] = wmma4(a, b, acc[nt]);
    }
  }
  // bias + relu -> LDS (reuse this wave's A-tile region; wave LDS ops are in-order)
  float* h1 = &At[wave][0][0];  // 16x64
#pragma unroll
  for (int r = 0; r < 8; ++r) {
    int row = r + 8 * half;  // C layout: VGPR r -> M=r (lanes 0-15), M=r+8 (16-31)
#pragma unroll
    for (int nt = 0; nt < 4; ++nt) {
      int col = nt * 16 + l15;
      h1[row * EDIM + col] = fmaxf(acc[nt][r] + b1[col], 0.f);
    }
  }

  // ---- layer 2: [16x64] @ [64x64] ----
  v8f acc2[4] = {v8zero(), v8zero(), v8zero(), v8zero()};
  for (int k0 = 0; k0 < EDIM; k0 += 4) {
    const int ka = k0 + 2 * half;
    v2f a;
    a.x = h1[l15 * EDIM + ka];
    a.y = h1[l15 * EDIM + ka + 1];
#pragma unroll
    for (int nt = 0; nt < 4; ++nt) {
      v2f b;
      b.x = W2[ka * EDIM + nt * 16 + l15];
      b.y = W2[(ka + 1) * EDIM + nt * 16 + l15];
      acc2[nt] = wmma4(a, b, acc2[nt]);
    }
  }
#pragma unroll
  for (int r = 0; r < 8; ++r) {
    int row = m0 + r + 8 * half;
    float di = dinv[row];
#pragma unroll
    for (int nt = 0; nt < 4; ++nt) {
      int col = nt * 16 + l15;
      float v = fmaxf(acc2[nt][r] + b2[col], 0.f);
      hout[(size_t)row * EDIM + col] = v;
      sout[(size_t)row * EDIM + col] = v * di;  // prescale for scatter
    }
  }
}

// ---------------------------------------------------------------------------
// edge scatter: agg[dst] += s[src]
// one wave32 per edge; lane owns features (2*lane, 2*lane+1) so the gather is
// a single contiguous 256B segment (global_load_b64 per lane).
// ---------------------------------------------------------------------------
__global__ void __launch_bounds__(256) k_scatter(
    const float* __restrict__ s, const int* __restrict__ esrc,
    const int* __restrict__ edst, float* __restrict__ agg, int E) {
  int wid = (blockIdx.x * blockDim.x + threadIdx.x) >> 5;
  int lane = threadIdx.x & 31;
  if (wid >= E) return;
  size_t so = (size_t)esrc[wid] * EDIM + 2 * lane;
  size_t dof = (size_t)edst[wid] * EDIM + 2 * lane;
  float2 v = *(const float2*)&s[so];
  atomicAdd(&agg[dof + 0], v.x);
  atomicAdd(&agg[dof + 1], v.y);
}

// Lh = h - agg*dinv ; s = Lh*dinv ; agg = 0 (reset fused for next scatter)
// float4-vectorized: 16 float4 per 64-float row.
__global__ void k_lap_post(const float* __restrict__ h, float* __restrict__ agg,
                           const float* __restrict__ dinv, float* __restrict__ Lh,
                           float* __restrict__ s, int n4) {
  int i = blockIdx.x * blockDim.x + threadIdx.x;
  if (i >= n4) return;
  float di = dinv[i >> 4];
  float4 h4 = ((const float4*)h)[i];
  float4 a4 = ((const float4*)agg)[i];
  ((float4*)agg)[i] = make_float4(0.f, 0.f, 0.f, 0.f);
  float4 lh, sv;
  lh.x = h4.x - a4.x * di; sv.x = lh.x * di;
  lh.y = h4.y - a4.y * di; sv.y = lh.y * di;
  lh.z = h4.z - a4.z * di; sv.z = lh.z * di;
  lh.w = h4.w - a4.w * di; sv.w = lh.w * di;
  ((float4*)Lh)[i] = lh;
  ((float4*)s)[i] = sv;
}

// h = t0*h + t1*Lh + t2*(Lh - agg*dinv) ; s = h*dinv ; agg = 0 (fused reset)
__global__ void k_poly(float* __restrict__ h, const float* __restrict__ Lh,
                       float* __restrict__ agg, const float* __restrict__ dinv,
                       float t0, float t1, float t2, float* __restrict__ s, int n4) {
  int i = blockIdx.x * blockDim.x + threadIdx.x;
  if (i >= n4) return;
  float di = dinv[i >> 4];
  float4 h4 = ((const float4*)h)[i];
  float4 l4 = ((const float4*)Lh)[i];
  float4 a4 = ((const float4*)agg)[i];
  ((float4*)agg)[i] = make_float4(0.f, 0.f, 0.f, 0.f);
  float4 hn, sv;
  hn.x = t0 * h4.x + t1 * l4.x + t2 * (l4.x - a4.x * di); sv.x = hn.x * di;
  hn.y = t0 * h4.y + t1 * l4.y + t2 * (l4.y - a4.y * di); sv.y = hn.y * di;
  hn.z = t0 * h4.z + t1 * l4.z + t2 * (l4.z - a4.z * di); sv.z = hn.z * di;
  hn.w = t0 * h4.w + t1 * l4.w + t2 * (l4.w - a4.w * di); sv.w = hn.w * di;
  ((float4*)h)[i] = hn;
  ((float4*)s)[i] = sv;
}

// ---------------------------------------------------------------------------
// final: combined = relu(feat[nodes]@weight + h[nodes]@W3W + b3w)
// ---------------------------------------------------------------------------
__global__ void __launch_bounds__(128) k_final(
    const float* __restrict__ feat, const float* __restrict__ hbuf,
    const float* __restrict__ weight, const float* __restrict__ W3W,
    const float* __restrict__ b3w, const int* __restrict__ nodes,
    float* __restrict__ out, int n) {
  __shared__ __align__(16) float FA[4][16][FDIM];  // 32 KB
  __shared__ __align__(16) float HA[4][16][EDIM];  // 16 KB
  const int wave = threadIdx.x >> 5;
  const int lane = threadIdx.x & 31;
  const int half = lane >> 4;
  const int l15 = lane & 15;
  const int m0 = (blockIdx.x * 4 + wave) * 16;
  const bool active = (m0 < n);  // n multiple of 16 (8192)

  if (active) {
#pragma unroll
    for (int i = 0; i < 16; ++i) {
      int idx = i * 32 + lane;
      int r = idx >> 5, c4 = idx & 31;
      int node = nodes[m0 + r];
      *(float4*)&FA[wave][r][c4 * 4] =
          ((const float4*)(feat + (size_t)node * FDIM))[c4];
    }
#pragma unroll
    for (int i = 0; i < 8; ++i) {
      int idx = i * 32 + lane;
      int r = idx >> 4, c4 = idx & 15;
      int node = nodes[m0 + r];
      *(float4*)&HA[wave][r][c4 * 4] =
          ((const float4*)(hbuf + (size_t)node * EDIM))[c4];
    }
  }
  __syncthreads();
  if (!active) return;

  v8f acc[4] = {v8zero(), v8zero(), v8zero(), v8zero()};
  // feat[nodes] @ weight  (K=128)
  for (int k0 = 0; k0 < FDIM; k0 += 4) {
    const int ka = k0 + 2 * half;
    v2f a;
    a.x = FA[wave][l15][ka];
    a.y = FA[wave][l15][ka + 1];
#pragma unroll
    for (int nt = 0; nt < 4; ++nt) {
      v2f b;
      b.x = weight[ka * EDIM + nt * 16 + l15];
      b.y = weight[(ka + 1) * EDIM + nt * 16 + l15];
      acc[nt] = wmma4(a, b, acc[nt]);
    }
  }
  // h[nodes] @ (W3@weight)  (K=64)
  for (int k0 = 0; k0 < EDIM; k0 += 4) {
    const int ka = k0 + 2 * half;
    v2f a;
    a.x = HA[wave][l15][ka];
    a.y = HA[wave][l15][ka + 1];
#pragma unroll
    for (int nt = 0; nt < 4; ++nt) {
      v2f b;
      b.x = W3W[ka * EDIM + nt * 16 + l15];
      b.y = W3W[(ka + 1) * EDIM + nt * 16 + l15];
      acc[nt] = wmma4(a, b, acc[nt]);
    }
  }
#pragma unroll
  for (int r = 0; r < 8; ++r) {
    int row = m0 + r + 8 * half;
#pragma unroll
    for (int nt = 0; nt < 4; ++nt) {
      int col = nt * 16 + l15;
      out[(size_t)row * EDIM + col] = fmaxf(acc[nt][r] + b3w[col], 0.f);
    }
  }
}

// ---------------------------------------------------------------------------
extern "C" void kernel_launch(void* const* d_in, const int* in_sizes, int n_in,
                              void* d_out, int out_size, void* d_ws, size_t ws_size,
                              hipStream_t stream) {
  (void)n_in; (void)out_size; (void)ws_size;
  const float* feat   = (const float*)d_in[0];
  const float* Wclf   = (const float*)d_in[1];
  const float* bclf   = (const float*)d_in[2];
  const float* W1     = (const float*)d_in[3];
  const float* b1     = (const float*)d_in[4];
  const float* W2     = (const float*)d_in[5];
  const float* b2     = (const float*)d_in[6];
  const float* W3     = (const float*)d_in[7];
  const float* b3     = (const float*)d_in[8];
  const float* weight = (const float*)d_in[9];
  const int*   nodes  = (const int*)d_in[10];
  // d_in[11] = r1_neighs: unused (AGG_WEIGHT[0]==0 -> branch is dead code)
  const int*   esrc   = (const int*)d_in[12];
  const int*   edst   = (const int*)d_in[13];

  const int N = in_sizes[0] / FDIM;   // 50000 nodes in the full graph
  const int n = in_sizes[10];         // 8192 batch nodes
  const int E = in_sizes[12];         // 1.6M edges
  const int total = N * EDIM;
  const int total4 = total / 4;

  // workspace layout (floats); total ~52 MB
  float* ws   = (float*)d_ws;
  float* hbuf = ws;                          // N*64
  float* sbuf = hbuf + (size_t)total;        // N*64
  float* Lh   = sbuf + (size_t)total;        // N*64
  float* agg  = Lh + (size_t)total;          // N*64
  float* dinv = agg + (size_t)total;         // N
  float* W3W  = dinv + N;                    // 64*64
  float* b3w  = W3W + EDIM * EDIM;           // 64

  float* outC = (float*)d_out;               // combined [n,64]
  float* outS = outC + (size_t)n * EDIM;     // center_scores [n,2]

  // degrees -> dinv
  k_zero<<<(N + 255) / 256, 256, 0, stream>>>(dinv, N);
  k_deg<<<(E + 255) / 256, 256, 0, stream>>>(edst, dinv, E);
  k_dinv<<<(N + 255) / 256, 256, 0, stream>>>(dinv, N);

  // folded projection weights
  k_w3w<<<1, 256, 0, stream>>>(W3, weight, b3, W3W, b3w);

  // classifier scores at the 8192 selected nodes only
  k_scores<<<(n + 7) / 8, 256, 0, stream>>>(feat, Wclf, bclf, nodes, outS, n);

  // MLP over all N rows (writes h and prescaled s = h*dinv)
  const int tiles = N / 16;  // 3125
  k_mlp<<<(tiles + 3) / 4, 128, 0, stream>>>(feat, W1, b1, W2, b2, dinv, hbuf,
                                             sbuf, N);

  // agg zeroed once; lap/poly kernels reset it after consuming
  k_zero4<<<(total4 + 255) / 256, 256, 0, stream>>>((float4*)agg, total4);

  // Beta-wavelet PolyConv chain: 3 iterations x 2 Laplacian applications
  const float TH[3][3] = {{3.f, -3.f, 0.75f}, {0.f, 3.f, -1.5f}, {0.f, 0.f, 0.75f}};
  const int eb = (E + 7) / 8;            // one wave per edge, 8 waves/block
  const int tb4 = (total4 + 255) / 256;  // float4 elementwise passes
  for (int it = 0; it < 3; ++it) {
    k_scatter<<<eb, 256, 0, stream>>>(sbuf, esrc, edst, agg, E);
    k_lap_post<<<tb4, 256, 0, stream>>>(hbuf, agg, dinv, Lh, sbuf, total4);
    k_scatter<<<eb, 256, 0, stream>>>(sbuf, esrc, edst, agg, E);
    k_poly<<<tb4, 256, 0, stream>>>(hbuf, Lh, agg, dinv, TH[it][0], TH[it][1],
                                    TH[it][2], sbuf, total4);
  }

  // final fused gather + GEMMs + relu
  const int ftiles = (n + 15) / 16;
  k_final<<<(ftiles + 3) / 4, 128, 0, stream>>>(feat, hbuf, weight, W3W, b3w,
                                                nodes, outC, n);
}